// DeformableSelfAttention_19207093748556
// MI455X (gfx1250) — compile-verified
//
#include <hip/hip_runtime.h>
#include <math.h>

// ---------------------------------------------------------------------------
// Deformable self-attention for MI455X (gfx1250, wave32).
// B=2, Lq=4096, S=16384, D=256, H=8, P=4  (fixed by reference setup_inputs).
//
// Pipeline:
//   P1/P2: pack W_off|W_attn and W_out into WMMA B-fragment order (coalesced)
//   K1:    logits = Q @ [W_off | W_attn] + bias   (WMMA f32 16x16x4, LDS A-tile)
//   K2:    softmax + bilinear gather from value   (wave-per-query, b128 loads)
//   K3:    out    = attn_out @ W_out + b_out      (WMMA f32 16x16x4, LDS A-tile)
// ---------------------------------------------------------------------------

typedef float v2f __attribute__((ext_vector_type(2)));
typedef float v8f __attribute__((ext_vector_type(8)));

// Sanity check must be device-pass only: the builtin is not visible to the
// host compilation pass of hipcc.
#if defined(__HIP_DEVICE_COMPILE__)
#if !__has_builtin(__builtin_amdgcn_wmma_f32_16x16x4_f32)
#error "gfx1250 f32 WMMA builtin missing on this toolchain"
#endif
#endif

#define DMODEL 256
#define NHEAD 8
#define NPTS 4
#define NSLOT (NHEAD * NPTS)   // 32
#define NLOG  (NSLOT * 3)      // 96 = 64 offsets + 32 attn logits
#define LDS_STRIDE 260         // 16x256 tile, +4 pad: b64 frag reads hit 16 banks

// ---------------------------------------------------------------------------
// Weight pre-pack: out pair idx = (t*64 + kb4)*32 + lane holds
//   { W[kb4*4 + khalf    ][t*16 + (lane&15)],
//     W[kb4*4 + khalf + 1][t*16 + (lane&15)] },  khalf = (lane>>4)*2
// i.e. exactly the per-lane B fragment for WMMA step kb4 of N-tile t.
// ---------------------------------------------------------------------------
__global__ __launch_bounds__(256) void pack_w_kernel(
    const float* __restrict__ W, int ldw, int ntiles, float* __restrict__ out)
{
    int idx = blockIdx.x * blockDim.x + threadIdx.x;   // fragment-pair index
    int total = ntiles * 64 * 32;
    if (idx >= total) return;
    int lane = idx & 31;
    int kb4  = (idx >> 5) & 63;
    int t    = idx >> 11;
    int k = kb4 * 4 + ((lane >> 4) << 1);
    int n = t * 16 + (lane & 15);
    out[idx * 2]     = W[(size_t)k * ldw + n];
    out[idx * 2 + 1] = W[(size_t)(k + 1) * ldw + n];
}

// ---------------------------------------------------------------------------
// Kernel 1: logits[q,0:64] = q@W_off + b_off ; logits[q,64:96] = q@W_attn + b_attn
// One block = 6 waves, one 16-query tile staged in LDS; wave nt owns N-tile nt.
// Fragment layouts (CDNA5 ISA §7.12.2):
//   A 16x4 f32 : M = lane&15; lanes 0-15 -> K={0,1}, lanes 16-31 -> K={2,3}
//   C/D        : vgpr r -> M = r + 8*(lane>=16), N = lane&15
// ---------------------------------------------------------------------------
__global__ __launch_bounds__(192) void logits_kernel(
    const float* __restrict__ Q,
    const float* __restrict__ Wpack,      // 6 tiles, B-fragment order
    const float* __restrict__ b_off, const float* __restrict__ b_attn,
    float* __restrict__ logits)
{
    __shared__ float tile[16 * LDS_STRIDE];
    const int tid  = threadIdx.x;
    const int row0 = blockIdx.x * 16;

    // stage 16x256 A tile (1024 float4)
    for (int f = tid; f < 1024; f += 192) {
        int r = f >> 6, c4 = (f & 63) << 2;
        *(float4*)&tile[r * LDS_STRIDE + c4] =
            *(const float4*)(Q + (size_t)(row0 + r) * DMODEL + c4);
    }
    __syncthreads();

    const int nt    = tid >> 5;            // wave index = N-tile, 0..5
    const int lane  = tid & 31;
    const int mloc  = lane & 15;
    const int khalf = (lane >> 4) << 1;

    const float* arow = &tile[mloc * LDS_STRIDE + khalf];
    const v2f*   bp   = (const v2f*)Wpack + (size_t)nt * 64 * 32 + lane;

#if defined(__HIP_DEVICE_COMPILE__)
    v8f c = {0.f, 0.f, 0.f, 0.f, 0.f, 0.f, 0.f, 0.f};
    #pragma unroll 4
    for (int kb4 = 0; kb4 < 64; ++kb4) {
        v2f a = *(const v2f*)&arow[kb4 * 4];   // ds_load_b64
        v2f b = bp[kb4 * 32];                  // coalesced global_load_b64
        c = __builtin_amdgcn_wmma_f32_16x16x4_f32(false, a, false, b,
                                                  (short)0, c, false, false);
    }

    const float bv = (nt < 4) ? b_off[nt * 16 + mloc] : b_attn[(nt - 4) * 16 + mloc];
    #pragma unroll
    for (int r = 0; r < 8; ++r) {
        int m = r + ((lane >> 4) << 3);
        logits[(size_t)(row0 + m) * NLOG + nt * 16 + mloc] = c[r] + bv;
    }
#else
    (void)arow; (void)bp; (void)logits; (void)b_off; (void)b_attn; (void)khalf;
#endif
}

// ---------------------------------------------------------------------------
// Kernel 2: per-query softmax over 32 slots + bilinear gather from value.
// One wave per query; slot params broadcast via __shfl, all lanes gather
// 8 dims each (2x b128) from the two value rows per slot. L2-bound phase.
// ---------------------------------------------------------------------------
__global__ __launch_bounds__(256) void sample_kernel(
    const float* __restrict__ logits,     // [BLq, 96]
    const float* __restrict__ refpts,     // [BLq, 2]
    const float* __restrict__ value,      // [B, S, 256]
    float* __restrict__ attn_out,         // [BLq, 256]
    int Lq, int S, int BLq)
{
    const int q = blockIdx.x * (blockDim.x >> 5) + (threadIdx.x >> 5);
    if (q >= BLq) return;
    const int lane = threadIdx.x & 31;
    const int b = q / Lq;

    // softmax over the 32 attention logits (lane s holds slot s)
    float logit = logits[(size_t)q * NLOG + 2 * NSLOT + lane];
    float mx = logit;
    #pragma unroll
    for (int o = 16; o > 0; o >>= 1) mx = fmaxf(mx, __shfl_xor(mx, o, 32));
    float e = expf(logit - mx);
    float se = e;
    #pragma unroll
    for (int o = 16; o > 0; o >>= 1) se += __shfl_xor(se, o, 32);
    const float w = e / se;

    // sampling position for slot `lane` (grid_sample y-axis, align_corners)
    const float offy = logits[(size_t)q * NLOG + 2 * lane + 1];
    const float refy = refpts[(size_t)q * 2 + 1];
    const float pos  = (refy + offy + 1.0f) * 0.5f * (float)(S - 1);
    const float f0   = floorf(pos);
    const int   i0   = (int)f0;
    const float w1   = pos - f0;
    const int   i1   = i0 + 1;
    const float v0ok = (i0 >= 0 && i0 <= S - 1) ? 1.0f : 0.0f;   // zeros padding
    const float v1ok = (i1 >= 0 && i1 <= S - 1) ? 1.0f : 0.0f;
    const float c0s  = w * (1.0f - w1) * v0ok;
    const float c1s  = w * w1 * v1ok;
    const int   i0c  = min(max(i0, 0), S - 1);
    const int   i1c  = min(max(i1, 0), S - 1);

    // gather & blend: lane covers dims [8*lane, 8*lane+8)
    const float* vb = value + (size_t)b * S * DMODEL;
    float4 acc0 = {0.f, 0.f, 0.f, 0.f};
    float4 acc1 = {0.f, 0.f, 0.f, 0.f};
    for (int s = 0; s < NSLOT; ++s) {
        const int   a0 = __shfl(i0c, s, 32);
        const int   a1 = __shfl(i1c, s, 32);
        const float c0 = __shfl(c0s, s, 32);
        const float c1 = __shfl(c1s, s, 32);
        const float4* r0 = (const float4*)(vb + (size_t)a0 * DMODEL);
        const float4* r1 = (const float4*)(vb + (size_t)a1 * DMODEL);
        float4 x0 = r0[2 * lane], x1 = r0[2 * lane + 1];
        float4 y0 = r1[2 * lane], y1 = r1[2 * lane + 1];
        acc0.x = fmaf(c0, x0.x, fmaf(c1, y0.x, acc0.x));
        acc0.y = fmaf(c0, x0.y, fmaf(c1, y0.y, acc0.y));
        acc0.z = fmaf(c0, x0.z, fmaf(c1, y0.z, acc0.z));
        acc0.w = fmaf(c0, x0.w, fmaf(c1, y0.w, acc0.w));
        acc1.x = fmaf(c0, x1.x, fmaf(c1, y1.x, acc1.x));
        acc1.y = fmaf(c0, x1.y, fmaf(c1, y1.y, acc1.y));
        acc1.z = fmaf(c0, x1.z, fmaf(c1, y1.z, acc1.z));
        acc1.w = fmaf(c0, x1.w, fmaf(c1, y1.w, acc1.w));
    }
    float4* dst = (float4*)(attn_out + (size_t)q * DMODEL);
    dst[2 * lane]     = acc0;
    dst[2 * lane + 1] = acc1;
}

// ---------------------------------------------------------------------------
// Kernel 3: out = attn_out @ W_out + b_out   ([BLq,256] x [256,256])
// One block = 16 waves; A tile staged in LDS once, wave nt owns N-tile nt.
// ---------------------------------------------------------------------------
__global__ __launch_bounds__(512) void out_gemm_kernel(
    const float* __restrict__ X,          // [BLq, 256]
    const float* __restrict__ Wpack,      // 16 tiles, B-fragment order
    const float* __restrict__ bias,       // [256]
    float* __restrict__ Y)                // [BLq, 256]
{
    __shared__ float tile[16 * LDS_STRIDE];
    const int tid  = threadIdx.x;
    const int row0 = blockIdx.x * 16;

    for (int f = tid; f < 1024; f += 512) {
        int r = f >> 6, c4 = (f & 63) << 2;
        *(float4*)&tile[r * LDS_STRIDE + c4] =
            *(const float4*)(X + (size_t)(row0 + r) * DMODEL + c4);
    }
    __syncthreads();

    const int nt    = tid >> 5;            // 0..15
    const int lane  = tid & 31;
    const int mloc  = lane & 15;
    const int khalf = (lane >> 4) << 1;
    const int n0    = nt * 16;

    const float* arow = &tile[mloc * LDS_STRIDE + khalf];
    const v2f*   bp   = (const v2f*)Wpack + (size_t)nt * 64 * 32 + lane;

#if defined(__HIP_DEVICE_COMPILE__)
    v8f c = {0.f, 0.f, 0.f, 0.f, 0.f, 0.f, 0.f, 0.f};
    #pragma unroll 4
    for (int kb4 = 0; kb4 < 64; ++kb4) {
        v2f a = *(const v2f*)&arow[kb4 * 4];   // ds_load_b64
        v2f b = bp[kb4 * 32];                  // coalesced global_load_b64
        c = __builtin_amdgcn_wmma_f32_16x16x4_f32(false, a, false, b,
                                                  (short)0, c, false, false);
    }

    const float bv = bias[n0 + mloc];
    #pragma unroll
    for (int r = 0; r < 8; ++r) {
        int m = r + ((lane >> 4) << 3);
        Y[(size_t)(row0 + m) * DMODEL + n0 + mloc] = c[r] + bv;
    }
#else
    (void)arow; (void)bp; (void)Y; (void)bias; (void)khalf; (void)n0;
#endif
}

// ---------------------------------------------------------------------------
extern "C" void kernel_launch(void* const* d_in, const int* in_sizes, int n_in,
                              void* d_out, int out_size, void* d_ws, size_t ws_size,
                              hipStream_t stream) {
    (void)n_in; (void)out_size; (void)ws_size;
    const float* Q      = (const float*)d_in[0];
    /* d_in[1] = key: unused by the reference computation */
    const float* V      = (const float*)d_in[2];
    const float* refpts = (const float*)d_in[3];
    const float* W_off  = (const float*)d_in[4];
    const float* b_off  = (const float*)d_in[5];
    const float* W_attn = (const float*)d_in[6];
    const float* b_attn = (const float*)d_in[7];
    const float* W_out  = (const float*)d_in[8];
    const float* b_out  = (const float*)d_in[9];
    float* out = (float*)d_out;

    const int BLq = in_sizes[0] / DMODEL;          // B * Lq = 8192
    const int B   = 2;                              // fixed by reference setup
    const int Lq  = BLq / B;
    const int S   = (in_sizes[2] / DMODEL) / B;     // 16384

    // workspace layout
    float* logits   = (float*)d_ws;                         // [BLq, 96]
    float* attn_out = logits + (size_t)BLq * NLOG;          // [BLq, 256]
    float* packOut  = attn_out + (size_t)BLq * DMODEL;      // 16*64*32*2 floats
    float* packLog  = packOut + 16 * 64 * 32 * 2;           // 6*64*32*2 floats

    // pre-pack weights into B-fragment order (tiny, L2-resident)
    pack_w_kernel<<<dim3((16 * 64 * 32 + 255) / 256), 256, 0, stream>>>(
        W_out, DMODEL, 16, packOut);
    pack_w_kernel<<<dim3((4 * 64 * 32 + 255) / 256), 256, 0, stream>>>(
        W_off, 2 * NSLOT, 4, packLog);
    pack_w_kernel<<<dim3((2 * 64 * 32 + 255) / 256), 256, 0, stream>>>(
        W_attn, NSLOT, 2, packLog + (size_t)4 * 64 * 32 * 2);

    logits_kernel<<<dim3(BLq / 16), 192, 0, stream>>>(
        Q, packLog, b_off, b_attn, logits);

    sample_kernel<<<dim3((BLq + 7) / 8), 256, 0, stream>>>(
        logits, refpts, V, attn_out, Lq, S, BLq);

    out_gemm_kernel<<<dim3(BLq / 16), 512, 0, stream>>>(
        attn_out, packOut, b_out, out);
}